// HGCNLayer_73873437491766
// MI455X (gfx1250) — compile-verified
//
#include <hip/hip_runtime.h>
#include <hip/hip_bf16.h>
#include <math.h>
#include <stdint.h>

// HGCN layer: logmap0 -> GEMM(WMMA f32, TDM-staged W) + bias -> scatter-mean -> expmap0/proj
// Target: gfx1250 (CDNA5, wave32)

typedef __attribute__((ext_vector_type(2))) float v2f;
typedef __attribute__((ext_vector_type(8))) float v8f;
typedef __attribute__((ext_vector_type(4))) unsigned int v4u;
typedef __attribute__((ext_vector_type(8))) int v8i;
typedef __attribute__((ext_vector_type(4))) int v4i;

#define DIN   128
#define DOUT  128
#define EPSF  1e-5f
#define TM    128          // rows per GEMM block
#define LDW   (DIN + 4)    // padded LDS row stride (floats) to spread banks

#if __has_builtin(__builtin_amdgcn_tensor_load_to_lds) && __has_builtin(__builtin_amdgcn_s_wait_tensorcnt)
#define HGCN_USE_TDM 1
#else
#define HGCN_USE_TDM 0
#endif

__device__ __forceinline__ float waveReduceSum(float v) {
#pragma unroll
    for (int off = 16; off > 0; off >>= 1)
        v += __shfl_xor(v, off, 32);
    return v;
}

// ---------------- K0: zero fill (agg + deg) ----------------
__global__ __launch_bounds__(256) void hgcn_zero_kernel(float* __restrict__ p, long long n) {
    long long i = (long long)blockIdx.x * 256 + threadIdx.x;
    if (i < n) p[i] = 0.0f;
}

// ---------------- K1: logmap0 scale per row ----------------
__global__ __launch_bounds__(256) void hgcn_logmap_scale_kernel(const float* __restrict__ x,
                                                                float* __restrict__ scale, int N) {
    int row  = blockIdx.x * 8 + (threadIdx.x >> 5);
    int lane = threadIdx.x & 31;
    if (row >= N) return;
    float4 p = ((const float4*)(x + (size_t)row * DIN))[lane];
    float s = p.x * p.x + p.y * p.y + p.z * p.z + p.w * p.w;
    s = waveReduceSum(s);
    float norm = fmaxf(sqrtf(s), EPSF);
    float arg  = fminf(norm, 1.0f - EPSF);
    float sc   = atanhf(arg) / norm;
    if (lane == 0) scale[row] = sc;
}

// ---------------- K2: fused scale + GEMM (h @ W^T + b) via WMMA f32 ----------------
__global__ __launch_bounds__(256) void hgcn_gemm_wmma_kernel(const float* __restrict__ x,
                                                             const float* __restrict__ scale,
                                                             const float* __restrict__ W,
                                                             const float* __restrict__ bias,
                                                             float* __restrict__ ht, int N) {
    __shared__ float lds_h[TM * LDW];
    __shared__ float lds_w[DOUT * LDW];

    const int tid    = threadIdx.x;
    const int wave   = tid >> 5;
    const int m_base = blockIdx.x * TM;

#if HGCN_USE_TDM
    // Wave 0 issues one TDM DMA for the whole W tile (128x128 f32) while the
    // other work (scaled h staging) proceeds. D# pad fields inject +4 floats
    // per 128-float row so the LDS image lands with stride LDW=132 (bank
    // conflict avoidance), matching the WMMA fragment loads below.
    if (wave == 0) {
        unsigned long long ga = (unsigned long long)(uintptr_t)W;
        unsigned lds_off = (unsigned)(uintptr_t)(&lds_w[0]);   // flat->LDS offset (low 32 bits)
        v4u g0;
        g0.x = 1u;                                             // count=1, no gather
        g0.y = lds_off;                                        // lds_addr
        g0.z = (unsigned)(ga & 0xFFFFFFFFull);                 // global_addr[31:0]
        g0.w = (unsigned)((ga >> 32) & 0x1FFFFFFull)           // global_addr[56:32]
             | (2u << 30);                                     // type=2 (image)
        v8i g1;
        g1[0] = (int)(0x00020000u       // data_size=2 -> 4 bytes
               | (1u << 20)             // pad_enable
               | (6u << 22)             // pad_interval: every 128 DWORDs
               | (3u << 25));           // pad_amount: 4 DWORDs
        g1[1] = (int)(128u << 16);      // tensor_dim0[15:0] = 128
        g1[2] = (int)(128u << 16);      // tensor_dim0 hi=0 | tensor_dim1[15:0] = 128
        g1[3] = (int)(128u << 16);      // tensor_dim1 hi=0 | tile_dim0 = 128
        g1[4] = 128;                    // tile_dim1 = 128, tile_dim2 = 0
        g1[5] = 128;                    // tensor_dim0_stride = 128 elements
        g1[6] = 0;
        g1[7] = 0;
        v4i gz4 = {0, 0, 0, 0};         // 2-D tensor: groups 2/3 unused
        v8i gz8 = {0, 0, 0, 0, 0, 0, 0, 0};
        __builtin_amdgcn_tensor_load_to_lds(g0, g1, gz4, gz4, gz8, 0);
    }
#else
    // Manual W staging fallback.
    for (int i = tid; i < DOUT * (DIN / 4); i += 256) {
        int n  = i >> 5;
        int k4 = (i & 31) << 2;
        float4 w4 = *(const float4*)(W + (size_t)n * DIN + k4);
        float* d = &lds_w[n * LDW + k4];
        d[0] = w4.x; d[1] = w4.y; d[2] = w4.z; d[3] = w4.w;
    }
#endif

    // Stage h tile = x * scale (logmap0 fused), zero-fill rows past N.
    for (int i = tid; i < TM * (DIN / 4); i += 256) {
        int r  = i >> 5;
        int k4 = (i & 31) << 2;
        int gr = m_base + r;
        float4 h4 = make_float4(0.f, 0.f, 0.f, 0.f);
        float  sc = 0.f;
        if (gr < N) {
            h4 = *(const float4*)(x + (size_t)gr * DIN + k4);
            sc = scale[gr];
        }
        float* d = &lds_h[r * LDW + k4];
        d[0] = h4.x * sc; d[1] = h4.y * sc; d[2] = h4.z * sc; d[3] = h4.w * sc;
    }

#if HGCN_USE_TDM
    if (wave == 0) __builtin_amdgcn_s_wait_tensorcnt(0);
#endif
    __syncthreads();

    const int lane  = tid & 31;
    const int m0    = wave * 16;       // 16 output rows per wave
    const int lhalf = lane >> 4;       // 0 or 1
    const int l15   = lane & 15;

    // C/D layout: VGPR v, lanes0-15 -> (M=v, N=lane); lanes16-31 -> (M=v+8, N=lane-16).
    v8f acc[8];
#pragma unroll
    for (int t = 0; t < 8; ++t) {
        float bv = bias[t * 16 + l15];
#pragma unroll
        for (int j = 0; j < 8; ++j) acc[t][j] = bv;
    }

#if __has_builtin(__builtin_amdgcn_wmma_f32_16x16x4_f32)
    // 32 K-steps of 4; A: lane half selects K pair, B[k][n] = W[n][k].
#pragma unroll 4
    for (int ks = 0; ks < 32; ++ks) {
        const int k0 = ks * 4 + 2 * lhalf;
        v2f a = *(const v2f*)&lds_h[(m0 + l15) * LDW + k0];
#pragma unroll
        for (int t = 0; t < 8; ++t) {
            v2f bb = *(const v2f*)&lds_w[(t * 16 + l15) * LDW + k0];
            acc[t] = __builtin_amdgcn_wmma_f32_16x16x4_f32(
                false, a, false, bb, (short)0, acc[t], false, false);
        }
    }
#else
#pragma unroll
    for (int t = 0; t < 8; ++t) {
#pragma unroll
        for (int v = 0; v < 8; ++v) {
            int r = m0 + v + 8 * lhalf;
            int c = t * 16 + l15;
            float s = acc[t][v];
            for (int k = 0; k < DIN; ++k) s += lds_h[r * LDW + k] * lds_w[c * LDW + k];
            acc[t][v] = s;
        }
    }
#endif

    // Store D.
#pragma unroll
    for (int t = 0; t < 8; ++t) {
#pragma unroll
        for (int v = 0; v < 8; ++v) {
            int gr = m_base + m0 + v + 8 * lhalf;
            if (gr < N) ht[(size_t)gr * DOUT + t * 16 + l15] = acc[t][v];
        }
    }
}

// ---------------- K3: edge scatter (atomic f32, targets are L2-resident) ----------------
__global__ __launch_bounds__(256) void hgcn_scatter_kernel(const int* __restrict__ ei,
                                                           const float* __restrict__ ht,
                                                           float* __restrict__ agg,
                                                           float* __restrict__ deg, int E) {
    long long t = (long long)blockIdx.x * 256 + threadIdx.x;
    int e = (int)(t >> 5);
    if (e >= E) return;
    int lane = (int)(t & 31);
    int src = ei[e];
    int dst = ei[(size_t)E + e];
    float4 hv = *(const float4*)(ht + (size_t)src * DOUT + lane * 4);
    float* ad = agg + (size_t)dst * DOUT + lane * 4;
    atomicAdd(ad + 0, hv.x);
    atomicAdd(ad + 1, hv.y);
    atomicAdd(ad + 2, hv.z);
    atomicAdd(ad + 3, hv.w);
    if (lane == 0) atomicAdd(deg + dst, 1.0f);
}

// ---------------- K4: h_agg = ht + agg/deg, then expmap0 + proj ----------------
__global__ __launch_bounds__(256) void hgcn_expmap_kernel(const float* __restrict__ ht,
                                                          const float* __restrict__ agg,
                                                          const float* __restrict__ deg,
                                                          float* __restrict__ out, int N) {
    int row  = blockIdx.x * 8 + (threadIdx.x >> 5);
    int lane = threadIdx.x & 31;
    if (row >= N) return;
    float inv = 1.0f / fmaxf(deg[row], 1.0f);
    float4 h4 = *(const float4*)(ht  + (size_t)row * DOUT + lane * 4);
    float4 a4 = *(const float4*)(agg + (size_t)row * DOUT + lane * 4);
    float4 v;
    v.x = h4.x + a4.x * inv;
    v.y = h4.y + a4.y * inv;
    v.z = h4.z + a4.z * inv;
    v.w = h4.w + a4.w * inv;
    float s2 = v.x * v.x + v.y * v.y + v.z * v.z + v.w * v.w;
    s2 = waveReduceSum(s2);
    float truenorm = sqrtf(s2);
    float nclamp   = fmaxf(truenorm, EPSF);
    float sc       = tanhf(nclamp) / nclamp;         // expmap0 scale
    float ynorm    = fmaxf(truenorm * sc, EPSF);     // ||y|| for proj
    float maxn     = 1.0f - EPSF;
    float psc      = (ynorm > maxn) ? (maxn / ynorm) : 1.0f;
    float f = sc * psc;
    float4 o;
    o.x = v.x * f; o.y = v.y * f; o.z = v.z * f; o.w = v.w * f;
    *(float4*)(out + (size_t)row * DOUT + lane * 4) = o;
}

extern "C" void kernel_launch(void* const* d_in, const int* in_sizes, int n_in,
                              void* d_out, int out_size, void* d_ws, size_t ws_size,
                              hipStream_t stream) {
    (void)n_in; (void)out_size; (void)ws_size;
    const float* x    = (const float*)d_in[0];
    const int*   ei   = (const int*)d_in[1];   // jax int64 request silently -> int32 (no x64)
    const float* W    = (const float*)d_in[2];
    const float* bias = (const float*)d_in[3];
    const int N = in_sizes[0] / DIN;
    const int E = in_sizes[1] / 2;

    float* ws    = (float*)d_ws;
    float* ht    = ws;                               // N*128
    float* agg   = ht  + (size_t)N * DOUT;           // N*128
    float* deg   = agg + (size_t)N * DOUT;           // N   (contiguous after agg)
    float* scale = deg + N;                          // N
    float* out   = (float*)d_out;

    long long zn = (long long)N * DOUT + N;          // agg + deg
    hgcn_zero_kernel<<<(unsigned)((zn + 255) / 256), 256, 0, stream>>>(agg, zn);
    hgcn_logmap_scale_kernel<<<(N + 7) / 8, 256, 0, stream>>>(x, scale, N);
    hgcn_gemm_wmma_kernel<<<(N + TM - 1) / TM, 256, 0, stream>>>(x, scale, W, bias, ht, N);
    long long st = (long long)E * 32;
    hgcn_scatter_kernel<<<(unsigned)((st + 255) / 256), 256, 0, stream>>>(ei, ht, agg, deg, E);
    hgcn_expmap_kernel<<<(N + 7) / 8, 256, 0, stream>>>(ht, agg, deg, out, N);
}